// KernelDistance_74972949119307
// MI455X (gfx1250) — compile-verified
//
#include <hip/hip_runtime.h>

typedef __attribute__((ext_vector_type(2))) float v2f;
typedef __attribute__((ext_vector_type(4))) float v4f;
typedef __attribute__((ext_vector_type(8))) float v8f;

#define TILE    128
#define DDIM    64
#define LSTRIDE 68   // padded row stride in floats: 68 mod 64 = 4 banks/row -> conflict-free b64 frag loads

__global__ __launch_bounds__(256)
void rbf_wmma_f32_kernel(const float* __restrict__ pts,
                         float* __restrict__ out,
                         int N) {
    __shared__ float Al[TILE * LSTRIDE];
    __shared__ float Bl[TILE * LSTRIDE];
    __shared__ float sqA[TILE];
    __shared__ float sqB[TILE];

    const int b   = blockIdx.z;
    const int i0  = blockIdx.y * TILE;
    const int j0  = blockIdx.x * TILE;
    const int tid = threadIdx.x;
    const int lane = tid & 31;
    const int w    = tid >> 5;          // wave id, 0..7

    const float* baseA = pts + ((size_t)b * N + i0) * DDIM;
    const float* baseB = pts + ((size_t)b * N + j0) * DDIM;

    // ---- cooperative tile load: 128 rows x 64 floats = 2048 float4 per tile ----
    // Regular-temporal loads: the 4 MiB points tensor is heavily reused across
    // workgroups (each row read by 32 column-tiles) and should stay in L2.
#pragma unroll
    for (int i = 0; i < 8; ++i) {
        int idx = tid + i * 256;        // 0..2047
        int row = idx >> 4;             // 0..127
        int q   = idx & 15;             // float4 slot within a row
        v4f a  = *(const v4f*)(baseA + row * DDIM + q * 4);
        v4f bb = *(const v4f*)(baseB + row * DDIM + q * 4);
        *(v4f*)(&Al[row * LSTRIDE + q * 4]) = a;
        *(v4f*)(&Bl[row * LSTRIDE + q * 4]) = bb;
    }
    __syncthreads();

    // ---- squared norms of each row (threads 0..127 -> A, 128..255 -> B) ----
    if (tid < TILE) {
        float s = 0.f;
#pragma unroll
        for (int d = 0; d < DDIM; ++d) { float x = Al[tid * LSTRIDE + d]; s += x * x; }
        sqA[tid] = s;
    } else {
        int r = tid - TILE;
        float s = 0.f;
#pragma unroll
        for (int d = 0; d < DDIM; ++d) { float x = Bl[r * LSTRIDE + d]; s += x * x; }
        sqB[r] = s;
    }
    __syncthreads();

    // ---- WMMA main loop: wave w computes rows [w*16, w*16+16) x all 128 cols ----
    // A frag (16x4 f32): lane l -> row (l&15), K offset 2*(l>>4); one b64 LDS load.
    // B frag (4x16 f32): lane l -> col (l&15), same K striping -> identical pattern on Bl.
    const int mrow = lane & 15;
    const int koff = (lane >> 4) * 2;

    v8f acc[8] = {};   // 8 column tiles x 8 VGPRs accumulator

#pragma unroll
    for (int kk = 0; kk < 16; ++kk) {            // 16 steps of K=4 cover D=64
        const int kbase = kk * 4 + koff;
        v2f afrag = *(const v2f*)(&Al[(w * 16 + mrow) * LSTRIDE + kbase]);
#pragma unroll
        for (int c = 0; c < 8; ++c) {
            v2f bfrag = *(const v2f*)(&Bl[(c * 16 + mrow) * LSTRIDE + kbase]);
            acc[c] = __builtin_amdgcn_wmma_f32_16x16x4_f32(
                         /*neg_a=*/false, afrag,
                         /*neg_b=*/false, bfrag,
                         /*c_mod=*/(short)0, acc[c],
                         /*reuse_a=*/false, /*reuse_b=*/false);
        }
    }

    // ---- epilogue: d2 = |xi|^2 + |xj|^2 - 2*dot, clamp, exp(-d2/2) = exp2(-d2*log2(e)/2) ----
    // 256 MiB output, written once, never re-read: use non-temporal stores so the
    // stream doesn't sweep the 192 MB L2 and evict the hot points tiles.
    const int colLane = lane & 15;
    const int rowHi   = (lane >> 4) * 8;         // C/D layout: VGPR v -> row v + 8*(lane>=16)
    float* outBase = out + (size_t)b * N * N;
    constexpr float kScale = 0.72134752044448170367f;  // log2(e) / 2

#pragma unroll
    for (int c = 0; c < 8; ++c) {
        const int col = j0 + c * 16 + colLane;
        const float sb = sqB[c * 16 + colLane];
#pragma unroll
        for (int v = 0; v < 8; ++v) {
            const int rowT = w * 16 + rowHi + v;
            float d2 = sqA[rowT] + sb - 2.0f * acc[c][v];
            d2 = fmaxf(d2, 0.0f);
            float kval = __builtin_amdgcn_exp2f(-d2 * kScale);
            __builtin_nontemporal_store(kval, &outBase[(size_t)(i0 + rowT) * N + col]);
        }
    }
}

extern "C" void kernel_launch(void* const* d_in, const int* in_sizes, int n_in,
                              void* d_out, int out_size, void* d_ws, size_t ws_size,
                              hipStream_t stream) {
    const float* pts = (const float*)d_in[0];
    float* out = (float*)d_out;
    const int N = 4096;      // in_sizes[0] == 4*4096*64
    dim3 grid(N / TILE, N / TILE, 4);
    rbf_wmma_f32_kernel<<<grid, 256, 0, stream>>>(pts, out, N);
}